// CustomConv2d_7000796692497
// MI455X (gfx1250) — compile-verified
//
#include <hip/hip_runtime.h>

typedef __attribute__((ext_vector_type(2))) float v2f;
typedef __attribute__((ext_vector_type(8))) float v8f;
typedef __attribute__((ext_vector_type(2))) int   v2i;

#define CIN   128
#define COUT  256
#define HIN   64
#define WIN   64
#define HOUT  62
#define WOUT  62
#define BATCH 32
#define KDIM  (CIN * 9)            // 1152
#define POS   (HOUT * WOUT)        // 3844
#define NPOS  (BATCH * POS)        // 123008
#define BM    128                  // Cout tile per workgroup
#define BN    64                   // position tile per workgroup
#define KS    32                   // K slab staged in LDS
#define NSLAB (KDIM / KS)          // 36

#define AS_FLOATS ((KS / 2) * BM * 2)   // 4096 floats = 16 KB per buffer
#define BS_FLOATS ((KS / 2) * BN * 2)   // 2048 floats =  8 KB per buffer

#if defined(__gfx1250__) && \
    __has_builtin(__builtin_amdgcn_global_load_async_to_lds_b64) && \
    __has_builtin(__builtin_amdgcn_global_load_async_to_lds_b32)
#define USE_ASYNC 1
#else
#define USE_ASYNC 0
#endif

#define GLOBAL_AS __attribute__((address_space(1)))
#define LDS_AS    __attribute__((address_space(3)))

#if USE_ASYNC
// Parameter types per hipcc diagnostic: b64 takes v2i* (global), v2i* (LDS);
// b32 takes int* (global), int* (LDS); then imm offset, imm cpol.
#define ASYNC_B64(gsrc, ldst) \
    __builtin_amdgcn_global_load_async_to_lds_b64( \
        (GLOBAL_AS v2i*)(gsrc), (LDS_AS v2i*)(ldst), 0, 0)
#define ASYNC_B32(gsrc, ldst) \
    __builtin_amdgcn_global_load_async_to_lds_b32( \
        (GLOBAL_AS int*)(gsrc), (LDS_AS int*)(ldst), 0, 0)
#endif

static __device__ inline void wait_async0() {
#if USE_ASYNC
#if __has_builtin(__builtin_amdgcn_s_wait_asynccnt)
    __builtin_amdgcn_s_wait_asynccnt(0);
#else
    asm volatile("s_wait_asynccnt 0x0" ::: "memory");
#endif
#endif
}

// Implicit-GEMM conv2d: C[M=Cout, N=B*Ho*Wo] = W[M, K=Cin*9] * im2col(X)[K, N]
// Wave = 32x32 C tile (2x2 WMMA sub-tiles), workgroup = 128x64, double-buffered LDS.
__global__ __launch_bounds__(256) void conv2d_wmma_f32(
    const float* __restrict__ x,   // [B, Cin, 64, 64]
    const float* __restrict__ w,   // [Cout, Cin, 3, 3] => [Cout, K]
    float* __restrict__ out)       // [B, Cout, 62, 62]
{
    // Pair-interleaved LDS slabs: As[buf][k2][m][k&1], Bs[buf][k2][n][k&1]
    __shared__ __align__(16) float As[2][AS_FLOATS];   // 32 KB
    __shared__ __align__(16) float Bs[2][BS_FLOATS];   // 16 KB

    const int tid  = threadIdx.x;
    const int lane = tid & 31;          // wave32
    const int wave = tid >> 5;          // 0..7
    const int half = lane >> 4;         // selects K / K+2 per ISA A/B layout
    const int l16  = lane & 15;

    const int cobase = blockIdx.y * BM;
    const int nbase  = blockIdx.x * BN;

    const int mw = (wave & 3) * 32;     // wave M tile base: 0,32,64,96
    const int nw = (wave >> 2) * 32;    // wave N tile base: 0,32

    // A staging: thread -> (row m, 16-wide k chunk) : 8 async b64 each
    const int am  = tid >> 1;           // 0..127
    const int akc = (tid & 1) * 16;     // 0 or 16
    // B staging: thread -> (one k row, 8 n cols) : 8 async b32 each
    const int bkk = tid >> 3;           // 0..31
    const int bns = (tid & 7) * 8;      // 0..56

    int xbase[8];
    #pragma unroll
    for (int j = 0; j < 8; ++j) {
        int n  = nbase + bns + j;       // NPOS % BN == 0
        int b  = n / POS;
        int p  = n - b * POS;
        int oh = p / WOUT;
        int ow = p - oh * WOUT;
        xbase[j] = ((b * CIN) * HIN + oh) * WIN + ow;
    }

    v8f acc00 = {}, acc01 = {}, acc10 = {}, acc11 = {};

    auto stage = [&](int buf, int k0) {
        // ---- A: weights, k-contiguous pairs -> pair-interleaved LDS
        const float* wp = w + (size_t)(cobase + am) * KDIM + k0 + akc;
        float* asb = &As[buf][0];
        #pragma unroll
        for (int j = 0; j < 8; ++j) {
            int k = akc + 2 * j;
            float* dst = &asb[((k >> 1) * BM + am) * 2];
#if USE_ASYNC
            ASYNC_B64(wp + 2 * j, dst);
#else
            v2f pr; pr.x = wp[2 * j]; pr.y = wp[2 * j + 1];
            *(v2f*)dst = pr;
#endif
        }
        // ---- B: im2col gather of x
        int gk = k0 + bkk;
        int ci = gk / 9;
        int r9 = gk - ci * 9;
        int kh = r9 / 3;
        int kw = r9 - kh * 3;
        int off = ci * (HIN * WIN) + kh * WIN + kw;
        float* bsb = &Bs[buf][0];
        #pragma unroll
        for (int j = 0; j < 8; ++j) {
            float* dst = &bsb[((bkk >> 1) * BN + (bns + j)) * 2 + (bkk & 1)];
#if USE_ASYNC
            ASYNC_B32(x + xbase[j] + off, dst);
#else
            *dst = x[xbase[j] + off];
#endif
        }
    };

    stage(0, 0);
    wait_async0();
    __syncthreads();

    for (int s = 0; s < NSLAB; ++s) {
        const int cur = s & 1;
        if (s + 1 < NSLAB) stage(cur ^ 1, (s + 1) * KS);  // overlaps WMMAs below

        const float* asb = &As[cur][0];
        const float* bsb = &Bs[cur][0];
        #pragma unroll
        for (int ks = 0; ks < KS; ks += 4) {
            int k2 = (ks >> 1) + half;
            v2f a0 = *(const v2f*)&asb[(k2 * BM + mw + l16) * 2];
            v2f a1 = *(const v2f*)&asb[(k2 * BM + mw + 16 + l16) * 2];
            v2f b0 = *(const v2f*)&bsb[(k2 * BN + nw + l16) * 2];
            v2f b1 = *(const v2f*)&bsb[(k2 * BN + nw + 16 + l16) * 2];
            acc00 = __builtin_amdgcn_wmma_f32_16x16x4_f32(false, a0, false, b0, (short)0, acc00, false, false);
            acc01 = __builtin_amdgcn_wmma_f32_16x16x4_f32(false, a0, false, b1, (short)0, acc01, false, false);
            acc10 = __builtin_amdgcn_wmma_f32_16x16x4_f32(false, a1, false, b0, (short)0, acc10, false, false);
            acc11 = __builtin_amdgcn_wmma_f32_16x16x4_f32(false, a1, false, b1, (short)0, acc11, false, false);
        }
        wait_async0();      // next slab fully landed in LDS
        __syncthreads();    // all waves' staging visible / buffer reusable
    }

    // Epilogue: D layout -> VGPR v holds M=v (lanes 0-15), M=v+8 (lanes 16-31)
    #pragma unroll
    for (int nt = 0; nt < 2; ++nt) {
        int n = nbase + nw + nt * 16 + l16;
        int b = n / POS;
        int p = n - b * POS;
        float* op = out + (size_t)(b * COUT + cobase + mw) * POS + p;
        v8f accA = (nt == 0) ? acc00 : acc01;   // M sub-tile +0
        v8f accB = (nt == 0) ? acc10 : acc11;   // M sub-tile +16
        #pragma unroll
        for (int v = 0; v < 8; ++v) {
            int co = v + 8 * half;
            op[(size_t)co * POS]        = accA[v];
            op[(size_t)(co + 16) * POS] = accB[v];
        }
    }
}

extern "C" void kernel_launch(void* const* d_in, const int* in_sizes, int n_in,
                              void* d_out, int out_size, void* d_ws, size_t ws_size,
                              hipStream_t stream) {
    const float* x = (const float*)d_in[0];   // 32*128*64*64
    const float* w = (const float*)d_in[1];   // 256*128*3*3
    float* out = (float*)d_out;               // 32*256*62*62

    dim3 grid(NPOS / BN, COUT / BM);          // (1922, 2)
    dim3 block(256);
    conv2d_wmma_f32<<<grid, block, 0, stream>>>(x, w, out);
}